// IMF_86053964742945
// MI455X (gfx1250) — compile-verified
//
#include <hip/hip_runtime.h>
#include <hip/hip_bf16.h>
#include <math.h>

// Problem constants (reference: x shape (4, 2048, 2048), RANK=16, NUM_ITERS=10,
// L2=(0,0), L1_RATIO=0 -> l1=l2=0, soft_thresholding is identity).
constexpr int BATCH = 4;
constexpr int M = 2048;
constexpr int N = 2048;
constexpr int R = 16;
constexpr int NUM_ITERS = 10;
constexpr int POWER_ITERS = 6;   // subspace iteration steps approximating truncated SVD init
constexpr float EPSV = 1e-16f;

typedef __attribute__((ext_vector_type(2))) float v2f;
typedef __attribute__((ext_vector_type(8))) float v8f;

// ---------------------------------------------------------------------------
// Deterministic pseudo-random init for the power-iteration starting basis.
// ---------------------------------------------------------------------------
__device__ __forceinline__ float hash_unit(unsigned s) {
  s ^= s >> 16; s *= 0x7feb352dU;
  s ^= s >> 15; s *= 0x846ca68bU;
  s ^= s >> 16;
  return (float)(s & 0xFFFFFFu) * (1.0f / 16777216.0f) - 0.5f;
}

__global__ void init_v_kernel(float* __restrict__ V, int n) {
  int i = blockIdx.x * blockDim.x + threadIdx.x;
  if (i < n) V[i] = hash_unit((unsigned)i * 2654435761u + 12345u);
}

// ---------------------------------------------------------------------------
// Modified Gram-Schmidt on W (rows x R, row-major), one block per batch.
// Deterministic tree reductions (no FP atomics). norms[b*R+r] = column norms.
// ---------------------------------------------------------------------------
__global__ void mgs_kernel(float* __restrict__ W, float* __restrict__ norms, int rows) {
  __shared__ float sh[256];
  __shared__ float dots[R];
  float* Wb = W + (size_t)blockIdx.x * rows * R;
  const int tid = threadIdx.x;
  for (int r = 0; r < R; ++r) {
    float part[R];
    #pragma unroll
    for (int j = 0; j < R; ++j) part[j] = 0.0f;
    for (int row = tid; row < rows; row += 256) {
      float wr = Wb[(size_t)row * R + r];
      for (int j = r; j < R; ++j) part[j] += wr * Wb[(size_t)row * R + j];
    }
    for (int j = r; j < R; ++j) {
      sh[tid] = part[j];
      __syncthreads();
      for (int s = 128; s > 0; s >>= 1) {
        if (tid < s) sh[tid] += sh[tid + s];
        __syncthreads();
      }
      if (tid == 0) dots[j] = sh[0];
      __syncthreads();
    }
    const float drr  = fmaxf(dots[r], 1e-30f);
    const float nrm  = sqrtf(drr);
    const float invd = 1.0f / drr;
    const float invn = 1.0f / nrm;
    if (tid == 0) norms[(size_t)blockIdx.x * R + r] = nrm;
    for (int row = tid; row < rows; row += 256) {
      float wr = Wb[(size_t)row * R + r];
      for (int j = r + 1; j < R; ++j)
        Wb[(size_t)row * R + j] -= (dots[j] * invd) * wr;
      Wb[(size_t)row * R + r] = wr * invn;
    }
    __syncthreads();
  }
}

// ---------------------------------------------------------------------------
// A = X @ V  (per batch: [M,N] @ [N,R] -> [M,R]).  One wave per 16-row M-tile.
// X rows are staged into LDS with coalesced float4 loads (X-row reads along K
// are otherwise badly strided), then consumed by V_WMMA_F32_16X16X4_F32.
// grid = BATCH*128/8 blocks of 256 threads (8 waves). EXEC all-ones everywhere.
// ---------------------------------------------------------------------------
constexpr int KC  = 128;
constexpr int KCP = KC + 4;  // pad LDS row stride to break bank conflicts

__global__ void gemm_xv_kernel(const float* __restrict__ X, const float* __restrict__ V,
                               float* __restrict__ A) {
  __shared__ float xs[8][16 * KCP];
  const int wid  = threadIdx.x >> 5;
  const int lane = threadIdx.x & 31;
  const int half = lane >> 4;
  const int lr   = lane & 15;
  const int gw    = blockIdx.x * 8 + wid;
  const int batch = gw >> 7;          // 128 M-tiles per batch
  const int mt    = gw & 127;
  const size_t m0 = (size_t)mt * 16;
  const float* Xb = X + (size_t)batch * M * N + m0 * N;
  const float* Vb = V + (size_t)batch * N * R;
  float* Ab       = A + (size_t)batch * M * R + m0 * R;

  v8f acc = {0.f, 0.f, 0.f, 0.f, 0.f, 0.f, 0.f, 0.f};
  for (int k0 = 0; k0 < N; k0 += KC) {
    // Stage 16 x KC tile of X (coalesced: 32 lanes x float4 per row).
    #pragma unroll
    for (int row = 0; row < 16; ++row) {
      const float4 d = *(const float4*)(Xb + (size_t)row * N + k0 + lane * 4);
      *(float4*)&xs[wid][row * KCP + lane * 4] = d;
    }
    for (int kk = 0; kk < KC; kk += 4) {
      // A-frag (16x4 f32): lane<16 -> K={kk,kk+1}, lane>=16 -> K={kk+2,kk+3}
      v2f a = *(const v2f*)&xs[wid][lr * KCP + kk + 2 * half];
      // B-frag (4x16): col = lr, VGPR0/1 = K slots per half-wave
      v2f b;
      const float* vp = Vb + (size_t)(k0 + kk + 2 * half) * R + lr;
      b.x = vp[0];
      b.y = vp[R];
      acc = __builtin_amdgcn_wmma_f32_16x16x4_f32(false, a, false, b,
                                                  (short)0, acc, false, false);
    }
  }
  #pragma unroll
  for (int j = 0; j < 8; ++j)
    Ab[(size_t)(j + 8 * half) * R + lr] = acc[j];   // D: VGPR j -> row j+8*half, col lr
}

// ---------------------------------------------------------------------------
// A = X^T @ U (per batch: [N,M]^T-view @ [M,R] -> [N,R]). A-frag loads are
// naturally coalesced (16 consecutive floats along a row of X), so no staging.
// ---------------------------------------------------------------------------
__global__ void gemm_xtv_kernel(const float* __restrict__ X, const float* __restrict__ U,
                                float* __restrict__ A) {
  const int wid  = threadIdx.x >> 5;
  const int lane = threadIdx.x & 31;
  const int half = lane >> 4;
  const int lr   = lane & 15;
  const int gw    = blockIdx.x * 8 + wid;
  const int batch = gw >> 7;
  const int nt    = gw & 127;
  const size_t n0 = (size_t)nt * 16;
  const float* Xb = X + (size_t)batch * M * N;
  const float* Ub = U + (size_t)batch * M * R;
  float* Ab       = A + (size_t)batch * N * R + n0 * R;

  v8f acc = {0.f, 0.f, 0.f, 0.f, 0.f, 0.f, 0.f, 0.f};
  for (int k = 0; k < M; k += 4) {
    v2f a, b;
    // A[row=lr, Kslot] = X^T[n0+lr, k+Kslot] = X[(k+Kslot)*N + n0+lr]
    a.x = Xb[(size_t)(k + 2 * half)     * N + n0 + lr];
    a.y = Xb[(size_t)(k + 2 * half + 1) * N + n0 + lr];
    const float* up = Ub + (size_t)(k + 2 * half) * R + lr;
    b.x = up[0];
    b.y = up[R];
    // Prefetch X rows ~16 K-steps ahead (stays inside the 192MB L2-resident X).
    if (k + 16 < M)
      __builtin_prefetch(Xb + (size_t)(k + 16 + 2 * half) * N + n0 + lr, 0, 1);
    acc = __builtin_amdgcn_wmma_f32_16x16x4_f32(false, a, false, b,
                                                (short)0, acc, false, false);
  }
  #pragma unroll
  for (int j = 0; j < 8; ++j)
    Ab[(size_t)(j + 8 * half) * R + lr] = acc[j];
}

// ---------------------------------------------------------------------------
// Bm = W^T W  (R x R Gram matrix). One wave per batch; A and B fragments are
// the identical loads (A = W^T so A[r, kslot] = W[kslot, r] = B[kslot, r]).
// ---------------------------------------------------------------------------
__global__ void gram_kernel(const float* __restrict__ W, float* __restrict__ Bm, int rows) {
  const int batch = blockIdx.x;
  const int lane  = threadIdx.x & 31;
  const int half  = lane >> 4;
  const int lr    = lane & 15;
  const float* Wb = W + (size_t)batch * rows * R;

  v8f acc = {0.f, 0.f, 0.f, 0.f, 0.f, 0.f, 0.f, 0.f};
  for (int k = 0; k < rows; k += 4) {
    v2f a;
    const float* wp = Wb + (size_t)(k + 2 * half) * R + lr;
    a.x = wp[0];
    a.y = wp[R];
    acc = __builtin_amdgcn_wmma_f32_16x16x4_f32(false, a, false, a,
                                                (short)0, acc, false, false);
  }
  #pragma unroll
  for (int j = 0; j < 8; ++j)
    Bm[(size_t)batch * (R * R) + (size_t)(j + 8 * half) * R + lr] = acc[j];
}

// ---------------------------------------------------------------------------
// W[i] = round(W[i] * sqrt(s[col]))  -- SVD-style init scaling + projection.
// rows-per-batch * R = 32768 = 2^15.
// ---------------------------------------------------------------------------
__global__ void scale_round_kernel(float* __restrict__ W, const float* __restrict__ snorm, int n) {
  int i = blockIdx.x * blockDim.x + threadIdx.x;
  if (i < n) {
    const int col   = i & (R - 1);
    const int batch = i >> 15;
    W[i] = rintf(W[i] * sqrtf(fmaxf(snorm[batch * R + col], 0.0f)));
  }
}

// ---------------------------------------------------------------------------
// One BCD sweep over the R columns of U (reference _update_u with l1=l2=0):
//   for r: term2 = sum_j u[j]*b[j][r] - u[r]*b[r][r]
//          u[r]  = rint((a[r]-term2 + EPS) / (b[r][r] + EPS))
// One thread per row; u row lives in registers across the sequential sweep.
// rintf == round-half-to-even == jnp.round.
// ---------------------------------------------------------------------------
__global__ void bcd_update_kernel(float* __restrict__ U, const float* __restrict__ A,
                                  const float* __restrict__ Bm) {
  __shared__ float bs[R * R];
  const int tid = threadIdx.x;
  const int gid = blockIdx.x * 256 + tid;        // global row id across batches
  const int batch = gid >> 11;                   // 2048 rows per batch
  bs[tid] = Bm[(size_t)batch * (R * R) + tid];
  __syncthreads();

  float u[R], a[R];
  float* up       = U + (size_t)gid * R;
  const float* ap = A + (size_t)gid * R;
  #pragma unroll
  for (int i = 0; i < R; ++i) { u[i] = up[i]; a[i] = ap[i]; }

  #pragma unroll
  for (int r = 0; r < R; ++r) {
    const float brr = bs[r * R + r];
    float t = 0.0f;
    #pragma unroll
    for (int j = 0; j < R; ++j) t += u[j] * bs[j * R + r];
    t -= u[r] * brr;
    const float num = a[r] - t;
    u[r] = rintf((num + EPSV) / (brr + EPSV));
  }
  #pragma unroll
  for (int i = 0; i < R; ++i) up[i] = u[i];
}

// ---------------------------------------------------------------------------
// out = U @ V^T  (per batch: [M,R] @ [R,N] -> [M,N]). K = R = 16 -> 4 WMMA
// steps per 16x16 tile. 65536 tiles total; this is the only HBM-heavy kernel
// (67 MB of stores), everything per-tile is L2/register resident.
// ---------------------------------------------------------------------------
__global__ void out_gemm_kernel(const float* __restrict__ U, const float* __restrict__ V,
                                float* __restrict__ out) {
  const int wid  = threadIdx.x >> 5;
  const int lane = threadIdx.x & 31;
  const int half = lane >> 4;
  const int lr   = lane & 15;
  const int gw    = blockIdx.x * 8 + wid;
  const int batch = gw >> 14;          // 128*128 tiles per batch
  const int rem   = gw & 16383;
  const size_t m0 = (size_t)(rem >> 7) * 16;
  const size_t n0 = (size_t)(rem & 127) * 16;
  const float* Ub = U + (size_t)batch * M * R;
  const float* Vb = V + (size_t)batch * N * R;
  float* Ob       = out + (size_t)batch * M * N;

  v8f acc = {0.f, 0.f, 0.f, 0.f, 0.f, 0.f, 0.f, 0.f};
  #pragma unroll
  for (int k = 0; k < R; k += 4) {
    // A[row=lr, Kslot] = U[m0+lr, k+Kslot]
    v2f a = *(const v2f*)(Ub + (m0 + lr) * R + k + 2 * half);
    // B[Kslot, col=lr] = V^T[k+Kslot, n0+lr] = V[n0+lr, k+Kslot]
    v2f b = *(const v2f*)(Vb + (n0 + lr) * R + k + 2 * half);
    acc = __builtin_amdgcn_wmma_f32_16x16x4_f32(false, a, false, b,
                                                (short)0, acc, false, false);
  }
  #pragma unroll
  for (int j = 0; j < 8; ++j)
    Ob[(m0 + j + 8 * half) * N + n0 + lr] = acc[j];
}

// ---------------------------------------------------------------------------
// Host-side pipeline (all launches on `stream`, graph-capture safe).
// Workspace layout (floats): U | V | Abuf | bmat | snorm  (~1.6 MB total).
// ---------------------------------------------------------------------------
extern "C" void kernel_launch(void* const* d_in, const int* in_sizes, int n_in,
                              void* d_out, int out_size, void* d_ws, size_t ws_size,
                              hipStream_t stream) {
  (void)in_sizes; (void)n_in; (void)out_size; (void)ws_size;
  const float* X = (const float*)d_in[0];
  float* out = (float*)d_out;

  float* ws    = (float*)d_ws;
  float* U     = ws;                       // BATCH*M*R
  float* V     = U + (size_t)BATCH * M * R;
  float* Abuf  = V + (size_t)BATCH * N * R;
  float* bmat  = Abuf + (size_t)BATCH * M * R;   // BATCH*R*R
  float* snorm = bmat + (size_t)BATCH * R * R;   // BATCH*R

  const int nVR = BATCH * N * R;
  const int nUR = BATCH * M * R;

  // ---- Rank-16 truncated-SVD init via block power iteration + MGS ----
  init_v_kernel<<<(nVR + 255) / 256, 256, 0, stream>>>(V, nVR);
  mgs_kernel<<<BATCH, 256, 0, stream>>>(V, snorm, N);
  for (int p = 0; p < POWER_ITERS; ++p) {
    gemm_xv_kernel<<<BATCH * (M / 16) / 8, 256, 0, stream>>>(X, V, U);
    mgs_kernel<<<BATCH, 256, 0, stream>>>(U, snorm, M);
    gemm_xtv_kernel<<<BATCH * (N / 16) / 8, 256, 0, stream>>>(X, U, V);
    mgs_kernel<<<BATCH, 256, 0, stream>>>(V, snorm, N);
  }
  // u = round(U * sqrt(s)), v = round(V * sqrt(s))
  scale_round_kernel<<<(nUR + 255) / 256, 256, 0, stream>>>(U, snorm, nUR);
  scale_round_kernel<<<(nVR + 255) / 256, 256, 0, stream>>>(V, snorm, nVR);

  // ---- 10 BCD iterations: u-sweep then v-sweep (l1 = l2 = 0) ----
  for (int it = 0; it < NUM_ITERS; ++it) {
    gram_kernel<<<BATCH, 32, 0, stream>>>(V, bmat, N);                       // b = V^T V
    gemm_xv_kernel<<<BATCH * (M / 16) / 8, 256, 0, stream>>>(X, V, Abuf);    // a = X V
    bcd_update_kernel<<<BATCH * M / 256, 256, 0, stream>>>(U, Abuf, bmat);

    gram_kernel<<<BATCH, 32, 0, stream>>>(U, bmat, M);                       // b = U^T U
    gemm_xtv_kernel<<<BATCH * (N / 16) / 8, 256, 0, stream>>>(X, U, Abuf);   // a = X^T U
    bcd_update_kernel<<<BATCH * N / 256, 256, 0, stream>>>(V, Abuf, bmat);
  }

  // ---- out = U @ V^T ----
  out_gemm_kernel<<<BATCH * (M / 16) * (N / 16) / 8, 256, 0, stream>>>(U, V, out);
}